// GProjection_70093866270806
// MI455X (gfx1250) — compile-verified
//
#include <hip/hip_runtime.h>
#include <hip/hip_bf16.h>

// ---------------------------------------------------------------------------
// GProjection for MI455X (gfx1250, wave32).
// Bandwidth-bound: 924MB coalesced f32 stores dominate; 63MB of feature maps
// live in the 192MB L2, so scattered bilinear corner gathers are L2-served.
// The 4-corner bilinear combine is done EXACTLY in f32 on the matrix pipe via
// V_WMMA_F32_16X16X4_F32 (K=4 == 4 corners): 16 rank-4 updates per
// 16-point x 16-channel tile, A masked to one point-row per WMMA.
// ---------------------------------------------------------------------------

typedef __attribute__((ext_vector_type(2))) float v2f;
typedef __attribute__((ext_vector_type(8))) float v8f;

#define GP_N     30000
#define GP_OUTC  963
#define CAM_SCALE (256.0f / 1920.0f)

__device__ __forceinline__ float gp_clamp(float x, float lo, float hi) {
    return fminf(fmaxf(x, lo), hi);
}

// One feature level: C channels, HxW spatial, output channel offset CHOFF.
// Wave handles 16 points (n0..n0+15). lane&15 = point id for projection state,
// and also the channel column for B/D tiles.
template <int C, int H, int W, int CHOFF>
__device__ __forceinline__ void sample_level(const float* __restrict__ feat,
                                             int b, int n0, float wn, float hn,
                                             int lane, float* __restrict__ out)
{
    const int  p  = lane & 15;
    const bool hi = lane >= 16;

    // grid_sample (align_corners=False): x = ((g+1)*W - 1)/2
    float fx = ((wn + 1.0f) * (float)W - 1.0f) * 0.5f;
    float fy = ((hn + 1.0f) * (float)H - 1.0f) * 0.5f;
    float x0f = floorf(fx), y0f = floorf(fy);
    int   x0 = (int)x0f,  y0 = (int)y0f;
    int   x1 = x0 + 1,    y1 = y0 + 1;
    float wx1 = fx - x0f, wx0 = 1.0f - wx1;
    float wy1 = fy - y0f, wy0 = 1.0f - wy1;

    const bool vx0 = (x0 >= 0) & (x0 < W);
    const bool vx1 = (x1 >= 0) & (x1 < W);
    const bool vy0 = (y0 >= 0) & (y0 < H);
    const bool vy1 = (y1 >= 0) & (y1 < H);

    // zeros padding folded into the weights (channel-independent => exact)
    float w00 = (vx0 & vy0) ? wx0 * wy0 : 0.0f;   // k=0 : (x0,y0)
    float w10 = (vx1 & vy0) ? wx1 * wy0 : 0.0f;   // k=1 : (x1,y0)
    float w01 = (vx0 & vy1) ? wx0 * wy1 : 0.0f;   // k=2 : (x0,y1)
    float w11 = (vx1 & vy1) ? wx1 * wy1 : 0.0f;   // k=3 : (x1,y1)

    const int cx0 = min(max(x0, 0), W - 1), cx1 = min(max(x1, 0), W - 1);
    const int cy0 = min(max(y0, 0), H - 1), cy1 = min(max(y1, 0), H - 1);
    const int sp00 = cy0 * W + cx0, sp10 = cy0 * W + cx1;
    const int sp01 = cy1 * W + cx0, sp11 = cy1 * W + cx1;

    // A-matrix layout (16x4 f32): lane l<16 holds M=l {K0,K1}; lane l+16 holds M=l {K2,K3}
    const float myA0 = hi ? w01 : w00;
    const float myA1 = hi ? w11 : w10;
    // matching corner spatial offsets this lane owns (for broadcasting into B)
    const int   myS0 = hi ? sp01 : sp00;
    const int   myS1 = hi ? sp11 : sp10;

    const size_t bbase = (size_t)b * C * (H * W);

    for (int t = 0; t < C / 16; ++t) {
        const int c = t * 16 + p;                         // channel column
        const float* __restrict__ fc = feat + bbase + (size_t)c * (H * W);

        v8f acc = {};
        #pragma unroll
        for (int i = 0; i < 16; ++i) {
            // B_i: point i's corners. B rows K0/K1 on lanes 0-15, K2/K3 on 16-31.
            const int src = (lane & 16) | i;              // lane half matches row half
            const int s0 = __shfl(myS0, src, 32);
            const int s1 = __shfl(myS1, src, 32);
            v2f bm;
            bm.x = fc[s0];                                // K=0 (lo) / K=2 (hi)
            bm.y = fc[s1];                                // K=1 (lo) / K=3 (hi)
            // A_i: weight matrix masked to row i (2 cndmasks)
            v2f am;
            am.x = (p == i) ? myA0 : 0.0f;
            am.y = (p == i) ? myA1 : 0.0f;
            acc = __builtin_amdgcn_wmma_f32_16x16x4_f32(
                /*neg_a=*/false, am, /*neg_b=*/false, bm,
                /*c_mod=*/(short)0, acc, /*reuse_a=*/false, /*reuse_b=*/false);
        }

        // D layout: VGPR j = row j (lanes 0-15) / row j+8 (lanes 16-31), col = lane&15.
        float* __restrict__ orow =
            out + (size_t)(b * GP_N + n0) * GP_OUTC + CHOFF + t * 16 + p;
        #pragma unroll
        for (int j = 0; j < 8; ++j) {
            const int m = j + (hi ? 8 : 0);
            orow[(size_t)m * GP_OUTC] = acc[j];
        }
    }
}

__global__ void __launch_bounds__(256)
gproj_kernel(const float* __restrict__ resolution,
             const float* __restrict__ feat0, const float* __restrict__ feat1,
             const float* __restrict__ feat2, const float* __restrict__ feat3,
             const float* __restrict__ inputs, const float* __restrict__ camK,
             float* __restrict__ out, int groups_per_b, int total_groups)
{
    const int wave = (int)((blockIdx.x * blockDim.x + threadIdx.x) >> 5);
    const int lane = (int)(threadIdx.x & 31);
    if (wave >= total_groups) return;                     // wave-uniform: EXEC stays all-1s

    const int b  = wave / groups_per_b;
    const int n0 = (wave % groups_per_b) * 16;
    const int p  = lane & 15;
    const int n  = n0 + p;

    // load this lane's point
    const size_t pbase = (size_t)(b * GP_N + n) * 3;
    const float in0 = inputs[pbase + 0];
    const float in1 = inputs[pbase + 1];
    const float in2 = inputs[pbase + 2];

    // pass-through channels 0..2 (lanes 0-15 only: one writer per point)
    if (lane < 16) {
        float* orow = out + (size_t)(b * GP_N + n) * GP_OUTC;
        orow[0] = in0; orow[1] = in1; orow[2] = in2;
    }

    // projection: pos = inputs + MESH_POS(0,0,-0.8); K = camK * (256/1920)
    const float h0 = (resolution[0] - 1.0f) * 0.5f;
    const float h1 = (resolution[1] - 1.0f) * 0.5f;
    const float K00 = camK[b * 9 + 0] * CAM_SCALE;
    const float K01 = camK[b * 9 + 1] * CAM_SCALE;
    const float K02 = camK[b * 9 + 2] * CAM_SCALE;
    const float K11 = camK[b * 9 + 4] * CAM_SCALE;
    const float K12 = camK[b * 9 + 5] * CAM_SCALE;

    const float X = in0, Y = in1, Z = in2 - 0.8f;
    const float wpix = (-K00 * X - K01 * Y) / Z + K02 - h0;
    const float hpix = K11 * (Y / Z) + K12 - h1;
    const float wn = gp_clamp(wpix / h0, -1.0f, 1.0f);
    const float hn = gp_clamp(hpix / h1, -1.0f, 1.0f);

    sample_level< 64, 128, 128,   3>(feat0, b, n0, wn, hn, lane, out);
    sample_level<128,  64,  64,  67>(feat1, b, n0, wn, hn, lane, out);
    sample_level<256,  32,  32, 195>(feat2, b, n0, wn, hn, lane, out);
    sample_level<512,  16,  16, 451>(feat3, b, n0, wn, hn, lane, out);
}

extern "C" void kernel_launch(void* const* d_in, const int* in_sizes, int n_in,
                              void* d_out, int out_size, void* d_ws, size_t ws_size,
                              hipStream_t stream)
{
    const float* resolution = (const float*)d_in[0];
    const float* feat0      = (const float*)d_in[1];
    const float* feat1      = (const float*)d_in[2];
    const float* feat2      = (const float*)d_in[3];
    const float* feat3      = (const float*)d_in[4];
    const float* inputs     = (const float*)d_in[5];
    const float* camK       = (const float*)d_in[6];
    float* out = (float*)d_out;

    const int B = in_sizes[6] / 9;                // 8
    const int N = in_sizes[5] / (3 * B);          // 30000 (divisible by 16)
    const int groups_per_b = N / 16;              // 1875
    const int total_groups = B * groups_per_b;    // 15000 waves
    const int threads = 256;                      // 8 waves/block
    const int blocks  = (total_groups * 32 + threads - 1) / threads;

    hipLaunchKernelGGL(gproj_kernel, dim3(blocks), dim3(threads), 0, stream,
                       resolution, feat0, feat1, feat2, feat3, inputs, camK,
                       out, groups_per_b, total_groups);
}